// ForwardRTUModel_4964982194384
// MI455X (gfx1250) — compile-verified
//
#include <hip/hip_runtime.h>
#include <hip/hip_bf16.h>

typedef __attribute__((ext_vector_type(2))) float v2f;
typedef __attribute__((ext_vector_type(4))) float v4f;
typedef __attribute__((ext_vector_type(8))) float v8f;

#define BN   64      // batch
#define DIN  512
#define HN   512

// Output offsets (floats), concatenated in reference return order.
#define OFF_H1  0
#define OFF_H2  (OFF_H1 + BN*HN)
#define OFF_N0  (OFF_H2 + BN*HN)
#define OFF_N1  (OFF_N0 + BN*HN)
#define OFF_N2  (OFF_N1 + BN*HN)
#define OFF_N3  (OFF_N2 + BN*HN)
#define OFF_N4  ((size_t)(OFF_N3 + BN*HN))
#define BIGSZ   ((size_t)BN*DIN*HN)
#define OFF_N5  (OFF_N4 + BIGSZ)
#define OFF_N6  (OFF_N4 + 2*BIGSZ)
#define OFF_N7  (OFF_N4 + 3*BIGSZ)

#define DPB 8    // d-rows per block in the streaming kernel

// ---------------------------------------------------------------------------
// Kernel 1: two f32 GEMMs via V_WMMA_F32_16X16X4_F32 + fused small elementwise
// outputs h_t_c1, h_t_c2, n0..n3. One wave per 16x16 output tile.
// ---------------------------------------------------------------------------
__global__ __launch_bounds__(128) void rtu_gemm_small(
    const float* __restrict__ x_t, const float* __restrict__ h_c1,
    const float* __restrict__ h_c2,
    const float* __restrict__ gm0, const float* __restrict__ gm1,
    const float* __restrict__ gm2, const float* __restrict__ gm3,
    const float* __restrict__ r_param, const float* __restrict__ theta_param,
    const float* __restrict__ wx1, const float* __restrict__ wx2,
    float* __restrict__ out)
{
    const int lane  = threadIdx.x & 31;
    const int wid   = blockIdx.x * 4 + (threadIdx.x >> 5); // 0..127
    const int tileM = wid & 3;     // 4 row tiles (B=64)
    const int tileN = wid >> 2;    // 32 col tiles (H=512)
    const int lmod  = lane & 15;
    const int lhi   = lane >> 4;

    const int rowA = tileM * 16 + lmod;   // M index for A fragment
    const int colB = tileN * 16 + lmod;   // N index for B fragment

    v8f c1 = {};
    v8f c2 = {};

    for (int k0 = 0; k0 < DIN; k0 += 4) {
        const int kA = k0 + 2 * lhi;
        // A 16x4: lane m=l&15 holds K pair {kA, kA+1} (contiguous in x_t row)
        v2f a = *(const v2f*)(x_t + rowA * DIN + kA);
        // B 4x16: lane n=l&15 holds K pair {kA, kA+1} (column of wx)
        v2f b1, b2;
        b1.x = wx1[kA * HN + colB];
        b1.y = wx1[(kA + 1) * HN + colB];
        b2.x = wx2[kA * HN + colB];
        b2.y = wx2[(kA + 1) * HN + colB];
        c1 = __builtin_amdgcn_wmma_f32_16x16x4_f32(false, a, false, b1,
                                                   (short)0, c1, false, false);
        c2 = __builtin_amdgcn_wmma_f32_16x16x4_f32(false, a, false, b2,
                                                   (short)0, c2, false, false);
    }

    // Per-column recurrence params: column h is fixed per lane.
    const int h = colB;
    const float rp = r_param[h];
    const float tp = theta_param[h];
    const float exp_r = __expf(rp);
    const float r     = __expf(-exp_r);
    const float theta = __expf(tp);
    const float g     = r * __cosf(theta);
    const float phi   = r * __sinf(theta);
    const float nrm   = __fsqrt_rn(1.0f - r * r);
    const float d_g_w_r     = -exp_r * g;
    const float d_phi_w_r   = -exp_r * phi;
    const float d_g_w_theta = -theta * phi;
    const float d_phi_w_theta = theta * g;
    const float d_norm_w_r  = exp_r * r * r / nrm;

    // C/D layout: VGPR v, lanes 0-15 -> row v; lanes 16-31 -> row v+8.
    #pragma unroll
    for (int v = 0; v < 8; ++v) {
        const int m   = tileM * 16 + v + 8 * lhi;
        const int idx = m * HN + h;
        const float w1 = c1[v];
        const float w2 = c2[v];
        const float hc1 = h_c1[idx];
        const float hc2 = h_c2[idx];
        const float g0 = gm0[idx], g1 = gm1[idx];
        const float g2 = gm2[idx], g3 = gm3[idx];

        out[OFF_H1 + idx] = g * hc1 - phi * hc2 + nrm * w1;
        out[OFF_H2 + idx] = g * hc2 + phi * hc1 + nrm * w2;
        out[OFF_N0 + idx] = d_g_w_r * hc1 + g * g0 - d_phi_w_r * hc2 - phi * g1
                            + d_norm_w_r * w1;
        out[OFF_N1 + idx] = d_g_w_r * hc2 + g * g1 + d_phi_w_r * hc1 + phi * g0
                            + d_norm_w_r * w2;
        out[OFF_N2 + idx] = d_g_w_theta * hc1 + g * g2 - d_phi_w_theta * hc2
                            - phi * g3;
        out[OFF_N3 + idx] = d_g_w_theta * hc2 + g * g3 + d_phi_w_theta * hc1
                            + phi * g2;
    }
}

// ---------------------------------------------------------------------------
// Kernel 2: streaming elementwise for n4..n7 (the ~540 MB of traffic).
// float4 per thread along H, nontemporal loads/stores (zero reuse data).
// ---------------------------------------------------------------------------
__global__ __launch_bounds__(128) void rtu_big(
    const float* __restrict__ x_t,
    const float* __restrict__ gm4, const float* __restrict__ gm5,
    const float* __restrict__ gm6, const float* __restrict__ gm7,
    const float* __restrict__ r_param, const float* __restrict__ theta_param,
    float* __restrict__ out)
{
    const int h  = threadIdx.x * 4;        // 128 threads cover H=512
    const int b  = blockIdx.y;
    const int d0 = blockIdx.x * DPB;

    const v4f rp = *(const v4f*)(r_param + h);
    const v4f tp = *(const v4f*)(theta_param + h);
    v4f g, phi, nrm;
    #pragma unroll
    for (int j = 0; j < 4; ++j) {
        const float exp_r = __expf(rp[j]);
        const float r     = __expf(-exp_r);
        const float theta = __expf(tp[j]);
        g[j]   = r * __cosf(theta);
        phi[j] = r * __sinf(theta);
        nrm[j] = __fsqrt_rn(1.0f - r * r);
    }

    float* __restrict__ n4o = out + OFF_N4;
    float* __restrict__ n5o = out + OFF_N5;
    float* __restrict__ n6o = out + OFF_N6;
    float* __restrict__ n7o = out + OFF_N7;

    #pragma unroll
    for (int i = 0; i < DPB; ++i) {
        const int d = d0 + i;
        const float x = x_t[b * DIN + d];                // uniform per block-row
        const size_t base = ((size_t)(b * DIN + d)) * HN + h;

        const v4f m4 = __builtin_nontemporal_load((const v4f*)(gm4 + base));
        const v4f m5 = __builtin_nontemporal_load((const v4f*)(gm5 + base));
        const v4f m6 = __builtin_nontemporal_load((const v4f*)(gm6 + base));
        const v4f m7 = __builtin_nontemporal_load((const v4f*)(gm7 + base));

        const v4f nx = nrm * x;
        __builtin_nontemporal_store(g * m4 - phi * m5 + nx, (v4f*)(n4o + base));
        __builtin_nontemporal_store(g * m5 + phi * m4,      (v4f*)(n5o + base));
        __builtin_nontemporal_store(g * m6 - phi * m7,      (v4f*)(n6o + base));
        __builtin_nontemporal_store(g * m7 + phi * m6 + nx, (v4f*)(n7o + base));
    }
}

// ---------------------------------------------------------------------------
extern "C" void kernel_launch(void* const* d_in, const int* in_sizes, int n_in,
                              void* d_out, int out_size, void* d_ws, size_t ws_size,
                              hipStream_t stream) {
    const float* x_t   = (const float*)d_in[0];
    const float* h_c1  = (const float*)d_in[1];
    const float* h_c2  = (const float*)d_in[2];
    const float* gm0   = (const float*)d_in[3];
    const float* gm1   = (const float*)d_in[4];
    const float* gm2   = (const float*)d_in[5];
    const float* gm3   = (const float*)d_in[6];
    const float* gm4   = (const float*)d_in[7];
    const float* gm5   = (const float*)d_in[8];
    const float* gm6   = (const float*)d_in[9];
    const float* gm7   = (const float*)d_in[10];
    const float* r_p   = (const float*)d_in[11];
    const float* th_p  = (const float*)d_in[12];
    const float* wx1   = (const float*)d_in[13];
    const float* wx2   = (const float*)d_in[14];
    float* out = (float*)d_out;

    // 128 output tiles of 16x16, 4 waves (128 threads) per block -> 32 blocks.
    rtu_gemm_small<<<32, 128, 0, stream>>>(x_t, h_c1, h_c2, gm0, gm1, gm2, gm3,
                                           r_p, th_p, wx1, wx2, out);

    // (DIN/DPB) x B blocks of 128 threads for the streaming part.
    rtu_big<<<dim3(DIN / DPB, BN), 128, 0, stream>>>(x_t, gm4, gm5, gm6, gm7,
                                                     r_p, th_p, out);
}